// FullQuantumModel_33638183862879
// MI455X (gfx1250) — compile-verified
//
#include <hip/hip_runtime.h>
#include <cstdint>

#define NQ 16
#define DIM 65536
#define HALF 32768
#define BATCH 128
#define NLAYERS 6

typedef float v2f __attribute__((ext_vector_type(2)));
typedef float v8f __attribute__((ext_vector_type(8)));
typedef unsigned int v4u __attribute__((ext_vector_type(4)));
typedef int v8i __attribute__((ext_vector_type(8)));
typedef int v4i __attribute__((ext_vector_type(4)));

// ---------------------------------------------------------------------------
// TDM descriptor for a 1-D contiguous tile: 32768 x 4B elements (128 KB).
// Group0: count=1 | lds_addr | global_addr | type=2
// Group1: data_size=2 (4B), tensor_dim0=32768, tensor_dim1=1, tile_dim0=32768,
//         tensor_dim0_stride=32768
// ---------------------------------------------------------------------------
__device__ __forceinline__ void tdm_desc(unsigned lds_off, const void* gaddr,
                                         v4u& g0, v8i& g1) {
  unsigned long long ga = (unsigned long long)gaddr;
  g0[0] = 1u;                                   // count=1
  g0[1] = lds_off;                              // lds_addr (bytes)
  g0[2] = (unsigned)ga;                         // global_addr[31:0]
  g0[3] = (unsigned)((ga >> 32) & 0x1FFFFFFull) | 0x80000000u; // addr[56:32] | type=2
  g1[0] = 0x20000;            // data_size=2 (4 bytes)
  g1[1] = (int)0x80000000;    // tensor_dim0[15:0]=0x8000 in [31:16]
  g1[2] = 0x10000;            // tensor_dim0 hi=0, tensor_dim1=1 in [31:16]
  g1[3] = (int)0x80000000;    // tile_dim0=0x8000 in [31:16]
  g1[4] = 0;                  // tile_dim1=0, tile_dim2=0 (1-D tile)
  g1[5] = 32768;              // tensor_dim0_stride lo
  g1[6] = 0;
  g1[7] = 0;
}

// ---------------------------------------------------------------------------
// Prep: per-wire fused U = RZ*RY*RX (2x2 complex) and per-layer 16x16 complex
// W = kron(U[w12],U[w13],U[w14],U[w15]) for the WMMA stage.
// Ug: [6][16][8] floats; Wg: [6][2][256] floats (Wr then Wi, row-major [k'][k])
// ---------------------------------------------------------------------------
__global__ void qsim_prep(const float* __restrict__ params,
                          float* __restrict__ Ug, float* __restrict__ Wg) {
  __shared__ float uS[16 * 8];
  const int l = blockIdx.x, t = threadIdx.x;
  if (t < 16) {
    const float* p = params + (l * NQ + t) * 3;
    float hx = 0.5f * p[0], hy = 0.5f * p[1], hz = 0.5f * p[2];
    float cx = cosf(hx), sx = sinf(hx);
    float cy = cosf(hy), sy = sinf(hy);
    float cz = cosf(hz), sz = sinf(hz);
    // M = RY @ RX
    float m00r = cy * cx, m00i = sy * sx;
    float m01r = -sy * cx, m01i = -cy * sx;
    float m10r = sy * cx, m10i = -cy * sx;
    float m11r = cy * cx, m11i = -sy * sx;
    // row0 *= (cz - i sz); row1 *= (cz + i sz)
    float u[8];
    u[0] = cz * m00r + sz * m00i;  u[1] = cz * m00i - sz * m00r;
    u[2] = cz * m01r + sz * m01i;  u[3] = cz * m01i - sz * m01r;
    u[4] = cz * m10r - sz * m10i;  u[5] = cz * m10i + sz * m10r;
    u[6] = cz * m11r - sz * m11i;  u[7] = cz * m11i + sz * m11r;
    for (int j = 0; j < 8; ++j) {
      uS[t * 8 + j] = u[j];
      Ug[(l * NQ + t) * 8 + j] = u[j];
    }
  }
  __syncthreads();
  const int kp = t >> 4, k = t & 15;
  float pr = 1.f, pi = 0.f;
#pragma unroll
  for (int q = 0; q < 4; ++q) {
    int w = 12 + q;                 // wire 12 <-> bit 3 of the low nibble
    int rb = (kp >> (3 - q)) & 1;
    int cb = (k >> (3 - q)) & 1;
    float er = uS[w * 8 + (rb * 2 + cb) * 2 + 0];
    float ei = uS[w * 8 + (rb * 2 + cb) * 2 + 1];
    float nr = pr * er - pi * ei;
    float ni = pr * ei + pi * er;
    pr = nr; pi = ni;
  }
  Wg[l * 512 + kp * 16 + k] = pr;
  Wg[l * 512 + 256 + kp * 16 + k] = pi;
}

// ---------------------------------------------------------------------------
// Init: real input -> complex SoA state, fused with layer-0 wire-0 gate
// (pairs differ in bit 15).
// ---------------------------------------------------------------------------
__global__ void qsim_init(const float* __restrict__ in, float* __restrict__ sRe,
                          float* __restrict__ sIm, const float* __restrict__ Ug) {
  int gid = blockIdx.x * 256 + threadIdx.x;       // 128*32768 threads
  int b = gid >> 15;
  int i = gid & (HALF - 1);
  const float* u = Ug;                            // layer 0, wire 0
  size_t iA = (size_t)b * DIM + i, iB = iA + HALF;
  float x0 = in[iA], x1 = in[iB];
  sRe[iA] = u[0] * x0 + u[2] * x1;
  sIm[iA] = u[1] * x0 + u[3] * x1;
  sRe[iB] = u[4] * x0 + u[6] * x1;
  sIm[iB] = u[5] * x0 + u[7] * x1;
}

// ---------------------------------------------------------------------------
// Local layer kernel: 256KB LDS chunk (bits 14..0 local, bit15 = h).
//  - TDM async load re/im planes to LDS
//  - butterflies for wires 1..11 (bits 14..4)
//  - WMMA f32 16x16x4: wires 12..15 as 16x16 complex GEMM (S' = S * W^T)
//  - whole CNOT chain (0,1)..(14,15) as one gather: src = d ^ (d>>1) ^ (h<<14)
//  - TDM async store back
// ---------------------------------------------------------------------------
__global__ __launch_bounds__(512) void qsim_local(
    float* __restrict__ sRe, float* __restrict__ sIm,
    const float* __restrict__ Ug, const float* __restrict__ Wg, int layer) {
  extern __shared__ float lds[];
  float* lre = lds;
  float* lim = lds + HALF;
  const int tid = threadIdx.x;
  const int b = blockIdx.x >> 1;
  const int h = blockIdx.x & 1;
  const size_t base = (size_t)b * DIM + (size_t)h * HALF;

  if (tid < 32) {                      // wave 0 drives the Tensor Data Mover
    v4u g0; v8i g1; v4i z4 = {0, 0, 0, 0}; v8i z8 = {0, 0, 0, 0, 0, 0, 0, 0};
    tdm_desc((unsigned)(size_t)lre, sRe + base, g0, g1);
    __builtin_amdgcn_tensor_load_to_lds(g0, g1, z4, z4, z8, 0);
    tdm_desc((unsigned)(size_t)lim, sIm + base, g0, g1);
    __builtin_amdgcn_tensor_load_to_lds(g0, g1, z4, z4, z8, 0);
    __builtin_amdgcn_s_wait_tensorcnt(0);
  }
  __syncthreads();

  // ---- 1q gates, wires 1..11 (bits 14..4) ----
  for (int w = 1; w <= 11; ++w) {
    const float* u = Ug + (layer * NQ + w) * 8;
    float u00r = u[0], u00i = u[1], u01r = u[2], u01i = u[3];
    float u10r = u[4], u10i = u[5], u11r = u[6], u11i = u[7];
    int bb = 15 - w;
    int low = (1 << bb) - 1;
#pragma unroll 4
    for (int p = 0; p < 32; ++p) {
      int pi = p * 512 + tid;                       // 16384 pairs
      int i0 = ((pi >> bb) << (bb + 1)) | (pi & low);
      int i1 = i0 | (1 << bb);
      float a0r = lre[i0], a0i = lim[i0], a1r = lre[i1], a1i = lim[i1];
      lre[i0] = u00r * a0r - u00i * a0i + u01r * a1r - u01i * a1i;
      lim[i0] = u00r * a0i + u00i * a0r + u01r * a1i + u01i * a1r;
      lre[i1] = u10r * a0r - u10i * a0i + u11r * a1r - u11i * a1i;
      lim[i1] = u10r * a0i + u10i * a0r + u11r * a1i + u11i * a1r;
    }
    __syncthreads();
  }

  // ---- wires 12..15 via WMMA: Sr' = Sr*Wr^T - Si*Wi^T ; Si' = Sr*Wi^T + Si*Wr^T
  {
    const float* Wr = Wg + layer * 512;
    const float* Wi = Wr + 256;
    const int lane = tid & 31, wave = tid >> 5;
    const int m = lane & 15, hi = lane >> 4;
    v2f Br[4], Bi[4], Bni[4];
#pragma unroll
    for (int ks = 0; ks < 4; ++ks) {
      int c = 4 * ks + 2 * hi;                 // B = W^T: lane holds Wr[m][c..c+1]
      Br[ks][0] = Wr[m * 16 + c];  Br[ks][1] = Wr[m * 16 + c + 1];
      Bi[ks][0] = Wi[m * 16 + c];  Bi[ks][1] = Wi[m * 16 + c + 1];
      Bni[ks][0] = -Bi[ks][0];     Bni[ks][1] = -Bi[ks][1];
    }
#pragma unroll
    for (int tt = 0; tt < 8; ++tt) {
      int tile = wave * 8 + tt;                // 16 waves x 8 = 128 tiles
      int tb = tile * 256;                     // 16 groups x 16 amplitudes
      v2f Ar[4], Ai[4];
#pragma unroll
      for (int ks = 0; ks < 4; ++ks) {
        int c = 4 * ks + 2 * hi;
        Ar[ks][0] = lre[tb + m * 16 + c];  Ar[ks][1] = lre[tb + m * 16 + c + 1];
        Ai[ks][0] = lim[tb + m * 16 + c];  Ai[ks][1] = lim[tb + m * 16 + c + 1];
      }
      v8f accR = {}; v8f accI = {};
#pragma unroll
      for (int ks = 0; ks < 4; ++ks)
        accR = __builtin_amdgcn_wmma_f32_16x16x4_f32(false, Ar[ks], false, Br[ks], (short)0, accR, false, false);
#pragma unroll
      for (int ks = 0; ks < 4; ++ks)
        accR = __builtin_amdgcn_wmma_f32_16x16x4_f32(false, Ai[ks], false, Bni[ks], (short)0, accR, false, false);
#pragma unroll
      for (int ks = 0; ks < 4; ++ks)
        accI = __builtin_amdgcn_wmma_f32_16x16x4_f32(false, Ar[ks], false, Bi[ks], (short)0, accI, false, false);
#pragma unroll
      for (int ks = 0; ks < 4; ++ks)
        accI = __builtin_amdgcn_wmma_f32_16x16x4_f32(false, Ai[ks], false, Br[ks], (short)0, accI, false, false);
#pragma unroll
      for (int j = 0; j < 8; ++j) {            // D layout: vgpr j -> row j + 8*hi
        int row = j + 8 * hi;
        lre[tb + row * 16 + m] = accR[j];
        lim[tb + row * 16 + m] = accI[j];
      }
    }
  }
  __syncthreads();

  // ---- CNOT chain (0,1)..(14,15) as a single gather permutation ----
  {
    float tmp[64];
#pragma unroll
    for (int p = 0; p < 64; ++p) {
      int d = p * 512 + tid;
      int s = (d ^ (d >> 1)) ^ (h << 14);
      tmp[p] = lre[s];
    }
    __syncthreads();
#pragma unroll
    for (int p = 0; p < 64; ++p) lre[p * 512 + tid] = tmp[p];
    __syncthreads();
#pragma unroll
    for (int p = 0; p < 64; ++p) {
      int d = p * 512 + tid;
      int s = (d ^ (d >> 1)) ^ (h << 14);
      tmp[p] = lim[s];
    }
    __syncthreads();
#pragma unroll
    for (int p = 0; p < 64; ++p) lim[p * 512 + tid] = tmp[p];
  }
  __syncthreads();

  if (tid < 32) {                      // TDM store back
    v4u g0; v8i g1; v4i z4 = {0, 0, 0, 0}; v8i z8 = {0, 0, 0, 0, 0, 0, 0, 0};
    tdm_desc((unsigned)(size_t)lre, sRe + base, g0, g1);
    __builtin_amdgcn_tensor_store_from_lds(g0, g1, z4, z4, z8, 0);
    tdm_desc((unsigned)(size_t)lim, sIm + base, g0, g1);
    __builtin_amdgcn_tensor_store_from_lds(g0, g1, z4, z4, z8, 0);
    __builtin_amdgcn_s_wait_tensorcnt(0);
  }
}

// ---------------------------------------------------------------------------
// Cross-half kernel: CNOT(15,0) of layer l fused with wire-0 gate of layer l+1.
// ---------------------------------------------------------------------------
__global__ void qsim_nonlocal(float* __restrict__ sRe, float* __restrict__ sIm,
                              const float* __restrict__ Ug, int nextLayer) {
  int gid = blockIdx.x * 256 + threadIdx.x;
  int b = gid >> 15;
  int i = gid & (HALF - 1);
  size_t iA = (size_t)b * DIM + i, iB = iA + HALF;
  float ar = sRe[iA], ai = sIm[iA], br = sRe[iB], bi = sIm[iB];
  if (i & 1) { float t; t = ar; ar = br; br = t; t = ai; ai = bi; bi = t; }
  const float* u = Ug + (nextLayer * NQ + 0) * 8;
  sRe[iA] = u[0] * ar - u[1] * ai + u[2] * br - u[3] * bi;
  sIm[iA] = u[0] * ai + u[1] * ar + u[2] * bi + u[3] * br;
  sRe[iB] = u[4] * ar - u[5] * ai + u[6] * br - u[7] * bi;
  sIm[iB] = u[4] * ai + u[5] * ar + u[6] * bi + u[7] * br;
}

// ---------------------------------------------------------------------------
// Final: CNOT(15,0) of layer 5 fused with |psi|^2 block reduction -> partials.
// Per block: b and bit14 are uniform. slot0 = class bit14, slot1 = class 2+bit14.
// ---------------------------------------------------------------------------
__global__ void qsim_final(const float* __restrict__ sRe,
                           const float* __restrict__ sIm,
                           float* __restrict__ partial) {
  __shared__ float s0[256], s1[256];
  int t = threadIdx.x, blk = blockIdx.x;
  int gid = blk * 256 + t;
  int b = gid >> 15;
  int i = gid & (HALF - 1);
  size_t iA = (size_t)b * DIM + i, iB = iA + HALF;
  float ar = sRe[iA], ai = sIm[iA], br = sRe[iB], bi = sIm[iB];
  if (i & 1) { float tt; tt = ar; ar = br; br = tt; tt = ai; ai = bi; bi = tt; }
  s0[t] = ar * ar + ai * ai;
  s1[t] = br * br + bi * bi;
  __syncthreads();
  for (int s = 128; s > 0; s >>= 1) {
    if (t < s) { s0[t] += s0[t + s]; s1[t] += s1[t + s]; }
    __syncthreads();
  }
  if (t == 0) { partial[blk * 2] = s0[0]; partial[blk * 2 + 1] = s1[0]; }
}

__global__ void qsim_reduce(const float* __restrict__ partial,
                            float* __restrict__ out) {
  int t = blockIdx.x * 256 + threadIdx.x;
  if (t >= BATCH * 4) return;
  int b = t >> 2, c = t & 3;
  int bit14 = c & 1, slot = c >> 1;
  float s = 0.f;
  for (int k = 0; k < 64; ++k)
    s += partial[(b * 128 + bit14 * 64 + k) * 2 + slot];
  out[t] = s;
}

// ---------------------------------------------------------------------------
extern "C" void kernel_launch(void* const* d_in, const int* in_sizes, int n_in,
                              void* d_out, int out_size, void* d_ws, size_t ws_size,
                              hipStream_t stream) {
  const float* state = (const float*)d_in[0];     // (128, 65536) f32
  const float* params = (const float*)d_in[1];    // (6, 16, 3) f32
  float* out = (float*)d_out;                     // (128, 4) f32
  float* ws = (float*)d_ws;

  float* sRe = ws;
  float* sIm = sRe + (size_t)BATCH * DIM;
  float* Ug = sIm + (size_t)BATCH * DIM;          // 6*16*8 floats
  float* Wg = Ug + NLAYERS * NQ * 8;              // 6*512 floats
  float* partial = Wg + NLAYERS * 512;            // 32768 floats

  const int shmem = 2 * HALF * (int)sizeof(float);  // 256 KB dynamic LDS
  (void)hipFuncSetAttribute(reinterpret_cast<const void*>(qsim_local),
                            hipFuncAttributeMaxDynamicSharedMemorySize, shmem);

  qsim_prep<<<NLAYERS, 256, 0, stream>>>(params, Ug, Wg);
  qsim_init<<<(BATCH * HALF) / 256, 256, 0, stream>>>(state, sRe, sIm, Ug);
  for (int l = 0; l < NLAYERS; ++l) {
    qsim_local<<<BATCH * 2, 512, shmem, stream>>>(sRe, sIm, Ug, Wg, l);
    if (l + 1 < NLAYERS)
      qsim_nonlocal<<<(BATCH * HALF) / 256, 256, 0, stream>>>(sRe, sIm, Ug, l + 1);
  }
  qsim_final<<<(BATCH * HALF) / 256, 256, 0, stream>>>(sRe, sIm, partial);
  qsim_reduce<<<2, 256, 0, stream>>>(partial, out);
}